// LearnedAggregationLayer_23141283791239
// MI455X (gfx1250) — compile-verified
//
#include <hip/hip_runtime.h>

#define Bq 16
#define Nq 8192
#define Cq 512
#define Hq 8
#define HDq 64
#define NCHUNK 32          // chunks per batch element
#define CHUNK 256          // tokens per block
#define SUB 64             // tokens staged in LDS per iteration
#define NSUB (CHUNK / SUB)
#define XROW 516           // 512 + 2-dword pad every 256 dwords (TDM pad_enable)
#define PROW 68            // Pm row pad (bank conflicts)

typedef __attribute__((ext_vector_type(2))) float v2f;
typedef __attribute__((ext_vector_type(8))) float v8f;
typedef __attribute__((ext_vector_type(4))) unsigned int v4u;
typedef __attribute__((ext_vector_type(8))) int v8i;
typedef __attribute__((ext_vector_type(4))) int v4i;

#if defined(__has_builtin)
#if __has_builtin(__builtin_amdgcn_tensor_load_to_lds)
#define HAVE_TDM 1
#endif
#endif
#ifndef HAVE_TDM
#define HAVE_TDM 0
#endif

__device__ __forceinline__ v8f wmma_f32(v2f a, v2f b, v8f c) {
  return __builtin_amdgcn_wmma_f32_16x16x4_f32(false, a, false, b, (short)0, c,
                                               false, false);
}

#if HAVE_TDM
// 1-D TDM load: 32768 f32 (one 64x512 token tile, contiguous in memory) into
// LDS at ldsAddr, inserting 2 pad DWORDs after every 256 DWORDs so the LDS
// image has row stride 516 floats (bank-conflict-free for the WMMA feeds).
__device__ __forceinline__ void tdm_load_tile(unsigned ldsAddr,
                                              const float* gsrc) {
  unsigned long long ga = (unsigned long long)(uintptr_t)gsrc;
  v4u g0;
  g0.x = 1u;                                   // count = 1 (valid descriptor)
  g0.y = ldsAddr;                              // lds_addr [63:32]
  g0.z = (unsigned)ga;                         // global_addr low
  g0.w = (unsigned)((ga >> 32) & 0x01FFFFFFull) | (2u << 30);  // type=2
  v8i g1;
  // data_size=4B(2), pad_enable, pad_interval=256dw(7), pad_amount=2dw(1)
  g1[0] = (int)((2u << 16) | (1u << 20) | (7u << 22) | (1u << 25));
  g1[1] = (int)0x80000000u;  // tensor_dim0 = 32768 (bits 63:48)
  g1[2] = (int)0x00010000u;  // tensor_dim0 hi=0, tensor_dim1 = 1
  g1[3] = (int)0x80000000u;  // tile_dim0 = 32768 (bits 127:112)
  g1[4] = 1;                 // tile_dim1 = 1
  g1[5] = 32768;             // tensor_dim0_stride
  g1[6] = 0;
  g1[7] = 0;
  v4i z4 = {0, 0, 0, 0};
  v8i z8 = {0, 0, 0, 0, 0, 0, 0, 0};
  __builtin_amdgcn_tensor_load_to_lds(g0, g1, z4, z4, z8, 0);
}
#endif

// LDS column index for x tiles: 2-dword gap after column 256
__device__ __forceinline__ int xidx(int c) { return c + ((c >> 8) << 1); }

// ---------------------------------------------------------------------------
// Kernel A: fold cls/Wq/Wk into W_logit (C x H)
// ---------------------------------------------------------------------------
__global__ void __launch_bounds__(256) prep_kernel(
    const float* __restrict__ cls, const float* __restrict__ Wq,
    const float* __restrict__ Wk, float* __restrict__ WL) {
  __shared__ float qs[Cq];
  const int tid = threadIdx.x;
  for (int i = tid; i < Cq; i += 256) {
    float acc = 0.f;
    for (int c = 0; c < Cq; ++c) acc += cls[c] * Wq[c * Cq + i];
    qs[i] = acc * 0.125f;  // HD^-0.5
  }
  __syncthreads();
  for (int i = tid; i < Cq * Hq; i += 256) {
    const int c = i >> 3, h = i & 7;
    float acc = 0.f;
    for (int d = 0; d < HDq; ++d)
      acc += Wk[c * Cq + h * HDq + d] * qs[h * HDq + d];
    WL[c * Hq + h] = acc;
  }
}

// ---------------------------------------------------------------------------
// Kernel B: fused flash pass with TDM double-buffered staging.
// One block = 256 tokens of one batch element -> partial (m, s, z[8][512]).
// ---------------------------------------------------------------------------
__global__ void __launch_bounds__(256) chunk_kernel(
    const float* __restrict__ x, const float* __restrict__ WLg,
    float* __restrict__ mbuf, float* __restrict__ sbuf,
    float* __restrict__ zbuf) {
  __shared__ float xs[2][SUB][XROW];   // 2 x 129 KB staged x tiles
  __shared__ float wl[Cq][16];         // 32 KB W_logit, cols 8..15 zero
  __shared__ float lg[2][SUB][Hq];     // logits (two K-half partials)
  __shared__ float Pm[16][PROW];       // exp(l-m), rows 8..15 stay zero
  __shared__ float mh[Hq], sh[Hq], fac[16];

  const int tid = threadIdx.x;
  const int lane = tid & 31;
  const int wave = tid >> 5;
  const int uwave = __builtin_amdgcn_readfirstlane((int)tid) >> 5;  // scalar
  const int b = blockIdx.x / NCHUNK;
  const int ck = blockIdx.x % NCHUNK;
  const long token0 = (long)ck * CHUNK;
  const float* xb = x + (long)b * Nq * Cq;

  // init LDS (zero-pad wl cols 8..15; zero Pm incl. pad)
  for (int i = tid; i < Cq * Hq; i += 256) {
    wl[i >> 3][i & 7] = WLg[i];
    wl[i >> 3][8 + (i & 7)] = 0.f;
  }
  for (int i = tid; i < 16 * PROW; i += 256) (&Pm[0][0])[i] = 0.f;
  if (tid < Hq) { mh[tid] = -__builtin_inff(); sh[tid] = 0.f; }
  if (tid < 16) fac[tid] = 1.f;

  v8f acc0 = {}, acc1 = {}, acc2 = {}, acc3 = {};  // z tiles (16h x 16col) x4

#if HAVE_TDM
  if (uwave == 0)
    tdm_load_tile((unsigned)(uintptr_t)&xs[0][0][0], xb + token0 * Cq);
#endif
  __syncthreads();

  int buf = 0;
  for (int sub = 0; sub < NSUB; ++sub) {
#if HAVE_TDM
    if (uwave == 0) {
      if (sub + 1 < NSUB) {
        tdm_load_tile((unsigned)(uintptr_t)&xs[buf ^ 1][0][0],
                      xb + (token0 + (long)(sub + 1) * SUB) * Cq);
        __builtin_amdgcn_s_wait_tensorcnt(1);  // tile `sub` done, next inflight
      } else {
        __builtin_amdgcn_s_wait_tensorcnt(0);
      }
    }
#else
    {  // fallback: cooperative float4 stage (float2 LDS stores: 8B aligned)
      const float4* xsrc = (const float4*)(xb + (token0 + (long)sub * SUB) * Cq);
      for (int i = tid; i < SUB * Cq / 4; i += 256) {
        const int row = i >> 7;
        const int c4 = (i & 127) * 4;
        float4 v = xsrc[i];
        float* dst = &xs[buf][row][xidx(c4)];
        *(float2*)(dst + 0) = make_float2(v.x, v.y);
        *(float2*)(dst + 2) = make_float2(v.z, v.w);
      }
    }
#endif
    __syncthreads();
    float (*xsb)[XROW] = xs[buf];

    // --- logits: wave w -> token tile (w&3), K-half (w>>2) ---
    {
      const int t = wave & 3;
      const int khalf = wave >> 2;
      const int m = lane & 15;
      const int ko = (lane >> 4) * 2;
      v8f d = {};
      const int k0 = khalf * (Cq / 2);
      for (int k = k0; k < k0 + Cq / 2; k += 4) {
        const int kc = k + ko;
        const int ka = xidx(kc);  // (kc,kc+1) never straddle the pad gap
        v2f a, bb;
        a.x = xsb[t * 16 + m][ka];
        a.y = xsb[t * 16 + m][ka + 1];
        bb.x = wl[kc][m];          // unconditional: cols 8..15 are zero
        bb.y = wl[kc + 1][m];
        d = wmma_f32(a, bb, d);
      }
      if (m < 8) {
        const int tokhi = (lane >> 4) * 8;
#pragma unroll
        for (int r = 0; r < 8; ++r)
          lg[khalf][t * 16 + r + tokhi][m] = d[r];
      }
    }
    __syncthreads();

    // --- online softmax stats per head ---
    if (tid < Hq) {
      const int h = tid;
      const float mo = mh[h];
      float msub = -__builtin_inff();
      for (int n = 0; n < SUB; ++n)
        msub = fmaxf(msub, lg[0][n][h] + lg[1][n][h]);
      const float mn = fmaxf(mo, msub);
      const float f = __expf(mo - mn);
      float s = 0.f;
      for (int n = 0; n < SUB; ++n) {
        const float p = __expf(lg[0][n][h] + lg[1][n][h] - mn);
        Pm[h][n] = p;
        s += p;
      }
      sh[h] = sh[h] * f + s;
      mh[h] = mn;
      fac[h] = f;
    }
    __syncthreads();

    // --- rescale accumulators, then z += P(16x64) @ x(64x512) via WMMA ---
    {
      const int m = lane & 15;
      const int ko = (lane >> 4) * 2;
      const int hhi = (lane >> 4) * 8;
      float fr[8];
#pragma unroll
      for (int r = 0; r < 8; ++r) fr[r] = fac[r + hhi];
#pragma unroll
      for (int r = 0; r < 8; ++r) {
        acc0[r] *= fr[r]; acc1[r] *= fr[r];
        acc2[r] *= fr[r]; acc3[r] *= fr[r];
      }
      // this wave's 4 column tiles all sit on one side of the 256 gap
      const int colA = wave * 64 + m + ((wave >= 4) ? 2 : 0);
      for (int k = 0; k < SUB; k += 4) {
        const int kc = k + ko;
        v2f a;
        a.x = Pm[m][kc];
        a.y = Pm[m][kc + 1];
        v2f b0, b1, b2, b3;
        b0.x = xsb[kc][colA +  0]; b0.y = xsb[kc + 1][colA +  0];
        b1.x = xsb[kc][colA + 16]; b1.y = xsb[kc + 1][colA + 16];
        b2.x = xsb[kc][colA + 32]; b2.y = xsb[kc + 1][colA + 32];
        b3.x = xsb[kc][colA + 48]; b3.y = xsb[kc + 1][colA + 48];
        acc0 = wmma_f32(a, b0, acc0);
        acc1 = wmma_f32(a, b1, acc1);
        acc2 = wmma_f32(a, b2, acc2);
        acc3 = wmma_f32(a, b3, acc3);
      }
    }
    __syncthreads();
    buf ^= 1;
  }

  // --- store per-chunk partials (heads 0..7 live in lanes 0..15) ---
  const long zbase = (long)(b * NCHUNK + ck) * Hq * Cq;
  if ((lane >> 4) == 0) {
    const int col0 = wave * 64 + lane;
#pragma unroll
    for (int r = 0; r < 8; ++r) {
      zbuf[zbase + r * Cq + col0 +  0] = acc0[r];
      zbuf[zbase + r * Cq + col0 + 16] = acc1[r];
      zbuf[zbase + r * Cq + col0 + 32] = acc2[r];
      zbuf[zbase + r * Cq + col0 + 48] = acc3[r];
    }
  }
  if (tid < Hq) {
    mbuf[(b * NCHUNK + ck) * Hq + tid] = mh[tid];
    sbuf[(b * NCHUNK + ck) * Hq + tid] = sh[tid];
  }
}

// ---------------------------------------------------------------------------
// Kernel C: combine chunk partials, apply Wv head projection -> pooled (B,C)
// ---------------------------------------------------------------------------
__global__ void __launch_bounds__(256) combine_kernel(
    const float* __restrict__ mbuf, const float* __restrict__ sbuf,
    const float* __restrict__ zbuf, const float* __restrict__ Wv,
    float* __restrict__ pooled) {
  __shared__ float redm[NCHUNK];
  __shared__ float e[NCHUNK];
  __shared__ float zv[Cq];
  __shared__ float Ssh;
  const int b = blockIdx.x >> 3;
  const int h = blockIdx.x & 7;
  const int tid = threadIdx.x;

  if (tid < NCHUNK) redm[tid] = mbuf[(b * NCHUNK + tid) * Hq + h];
  __syncthreads();
  if (tid == 0) {
    float mm = -__builtin_inff();
    for (int c = 0; c < NCHUNK; ++c) mm = fmaxf(mm, redm[c]);
    float S = 0.f;
    for (int c = 0; c < NCHUNK; ++c) {
      const float ee = __expf(redm[c] - mm);
      e[c] = ee;
      S += sbuf[(b * NCHUNK + c) * Hq + h] * ee;
    }
    Ssh = S;
  }
  __syncthreads();
  const float invS = 1.f / Ssh;
  for (int col = tid; col < Cq; col += 256) {
    float acc = 0.f;
    for (int c = 0; c < NCHUNK; ++c)
      acc += zbuf[((long)(b * NCHUNK + c) * Hq + h) * Cq + col] * e[c];
    zv[col] = acc * invS;
  }
  __syncthreads();
  for (int d = tid; d < HDq; d += 256) {
    float acc = 0.f;
    for (int c = 0; c < Cq; ++c) acc += zv[c] * Wv[c * Cq + h * HDq + d];
    pooled[(b * Hq + h) * HDq + d] = acc;
  }
}

// ---------------------------------------------------------------------------
// Kernel D: out = pooled @ Wp + bp
// ---------------------------------------------------------------------------
__global__ void __launch_bounds__(256) out_kernel(
    const float* __restrict__ pooled, const float* __restrict__ Wp,
    const float* __restrict__ bp, float* __restrict__ out) {
  __shared__ float pv[Cq];
  const int b = blockIdx.x;
  const int tid = threadIdx.x;
  for (int i = tid; i < Cq; i += 256) pv[i] = pooled[b * Cq + i];
  __syncthreads();
  for (int col = tid; col < Cq; col += 256) {
    float acc = bp[col];
    for (int j = 0; j < Cq; ++j) acc += pv[j] * Wp[j * Cq + col];
    out[b * Cq + col] = acc;
  }
}

extern "C" void kernel_launch(void* const* d_in, const int* in_sizes, int n_in,
                              void* d_out, int out_size, void* d_ws,
                              size_t ws_size, hipStream_t stream) {
  (void)in_sizes; (void)n_in; (void)out_size; (void)ws_size;
  const float* x   = (const float*)d_in[0];
  const float* cls = (const float*)d_in[1];
  const float* Wq  = (const float*)d_in[2];
  const float* Wk  = (const float*)d_in[3];
  const float* Wv  = (const float*)d_in[4];
  const float* Wp  = (const float*)d_in[5];
  const float* bp  = (const float*)d_in[6];
  float* out = (float*)d_out;

  float* ws = (float*)d_ws;
  float* WL     = ws;                       // 512*8
  float* mbuf   = ws + 4096;                // B*NCHUNK*H
  float* sbuf   = ws + 8192;                // B*NCHUNK*H
  float* zbuf   = ws + 12288;               // B*NCHUNK*H*C  (8 MB)
  float* pooled = zbuf + (long)Bq * NCHUNK * Hq * Cq;  // B*C

  prep_kernel<<<1, 256, 0, stream>>>(cls, Wq, Wk, WL);
  chunk_kernel<<<Bq * NCHUNK, 256, 0, stream>>>(x, WL, mbuf, sbuf, zbuf);
  combine_kernel<<<Bq * Hq, 256, 0, stream>>>(mbuf, sbuf, zbuf, Wv, pooled);
  out_kernel<<<Bq, 256, 0, stream>>>(pooled, Wp, bp, out);
}